// SignalReconstructor_36447092474548
// MI455X (gfx1250) — compile-verified
//
#include <hip/hip_runtime.h>
#include <math.h>

#define B_DIM 8192
#define C_DIM 64
#define EPS 1e-5f

typedef __attribute__((ext_vector_type(2))) float v2f;
typedef __attribute__((ext_vector_type(8))) float v8f;

// ---------------------------------------------------------------------------
// Kernel 1: build data = cat([pred, imfs], dim=1) and per-block partial sums
// ---------------------------------------------------------------------------
__global__ __launch_bounds__(256) void k_build_data(
    const float* __restrict__ pred, const float* __restrict__ imfs,
    float* __restrict__ data_out, float* __restrict__ partials)
{
    __shared__ float s_sum[256];
    __shared__ float s_sq[256];
    const int tid = threadIdx.x;
    const int idx = blockIdx.x * 256 + tid;          // 0 .. B*C-1
    const int c = idx & 63;
    const int b = idx >> 6;
    const float v = (c == 0) ? pred[b] : imfs[b * 63 + (c - 1)];
    data_out[idx] = v;                               // re-read by GEMM: keep RT
    s_sum[tid] = v;
    s_sq[tid]  = v * v;
    __syncthreads();
    #pragma unroll
    for (int off = 128; off > 0; off >>= 1) {
        if (tid < off) {
            s_sum[tid] += s_sum[tid + off];
            s_sq[tid]  += s_sq[tid + off];
        }
        __syncthreads();
    }
    if (tid == 0) {
        partials[blockIdx.x * 2 + 0] = s_sum[0];
        partials[blockIdx.x * 2 + 1] = s_sq[0];
    }
}

// ---------------------------------------------------------------------------
// Kernel 2: reduce 2048 partial pairs -> stats[0]=mu, stats[1]=std
// ---------------------------------------------------------------------------
__global__ __launch_bounds__(256) void k_stats(
    const float* __restrict__ partials, float* __restrict__ stats)
{
    __shared__ float s_sum[256];
    __shared__ float s_sq[256];
    const int tid = threadIdx.x;
    float a = 0.f, b = 0.f;
    for (int i = tid; i < 2048; i += 256) {
        a += partials[2 * i + 0];
        b += partials[2 * i + 1];
    }
    s_sum[tid] = a;
    s_sq[tid]  = b;
    __syncthreads();
    #pragma unroll
    for (int off = 128; off > 0; off >>= 1) {
        if (tid < off) {
            s_sum[tid] += s_sum[tid + off];
            s_sq[tid]  += s_sq[tid + off];
        }
        __syncthreads();
    }
    if (tid == 0) {
        const float n = (float)(B_DIM * C_DIM);
        const float mu = s_sum[0] / n;
        const float var = s_sq[0] / n - mu * mu;
        stats[0] = mu;
        stats[1] = sqrtf(var + EPS);
    }
}

// ---------------------------------------------------------------------------
// Kernel 3: per-row pipeline (RevIN norm -> MLP att -> conv/BN gate -> w)
// one thread per row; all weights staged in LDS; BN folded to scale/offset
// ---------------------------------------------------------------------------
__global__ __launch_bounds__(256) void k_rows(
    const float* __restrict__ data, const float* __restrict__ stats,
    const float* __restrict__ rev_w, const float* __restrict__ rev_b,
    const float* __restrict__ ca_w1, const float* __restrict__ ca_b1,
    const float* __restrict__ ca_w2, const float* __restrict__ ca_b2,
    const float* __restrict__ conv1_w, const float* __restrict__ conv1_b,
    const float* __restrict__ bn1_g, const float* __restrict__ bn1_b,
    const float* __restrict__ bn1_m, const float* __restrict__ bn1_v,
    const float* __restrict__ conv2_w, const float* __restrict__ conv2_b,
    const float* __restrict__ bn2_g, const float* __restrict__ bn2_b,
    const float* __restrict__ bn2_m, const float* __restrict__ bn2_v,
    float* __restrict__ w_out, float* __restrict__ xnorm_out)
{
    __shared__ float s_caw1[1024];   // (16,64)
    __shared__ float s_caw2[1024];   // (64,16)
    __shared__ float s_cw1[1024];    // (16,64) center tap
    __shared__ float s_cw2[1024];    // (64,16) center tap
    __shared__ float s_cab1[16], s_cb1[16], s_bn1s[16], s_bn1o[16];
    __shared__ float s_cab2[64], s_cb2[64], s_bn2s[64], s_bn2o[64];
    __shared__ float s_revw[64], s_revb[64];

    const int tid = threadIdx.x;
    for (int i = tid; i < 1024; i += 256) {
        s_caw1[i] = ca_w1[i];
        s_caw2[i] = ca_w2[i];
        s_cw1[i]  = conv1_w[i * 7 + 3];   // (o*64+ic)*7+3 flat == i*7+3
        s_cw2[i]  = conv2_w[i * 7 + 3];   // (o*16+ic)*7+3 flat == i*7+3
    }
    if (tid < 16) {
        s_cab1[tid] = ca_b1[tid];
        s_cb1[tid]  = conv1_b[tid];
        const float s = bn1_g[tid] * rsqrtf(bn1_v[tid] + EPS);
        s_bn1s[tid] = s;
        s_bn1o[tid] = bn1_b[tid] - bn1_m[tid] * s;
    }
    if (tid < 64) {
        s_cab2[tid] = ca_b2[tid];
        s_cb2[tid]  = conv2_b[tid];
        const float s = bn2_g[tid] * rsqrtf(bn2_v[tid] + EPS);
        s_bn2s[tid] = s;
        s_bn2o[tid] = bn2_b[tid] - bn2_m[tid] * s;
        s_revw[tid] = rev_w[tid];
        s_revb[tid] = rev_b[tid];
    }
    __syncthreads();

    const float mu = stats[0];
    const float sd = stats[1];
    const int row = blockIdx.x * 256 + tid;
    const float* __restrict__ dR = data + (size_t)row * 64;

    float a[64];   // x_norm, later x
    #pragma unroll
    for (int c = 0; c < 64; ++c) {
        const float xn = (dR[c] - mu) / sd * s_revw[c] + s_revb[c];
        a[c] = xn;
        // streamed output, never re-read on device -> non-temporal
        __builtin_nontemporal_store(xn, &xnorm_out[(size_t)row * 64 + c]);
    }

    // channel attention MLP: 64 -> 16 (relu) -> 64
    float t1[16];
    #pragma unroll
    for (int j = 0; j < 16; ++j) {
        float acc = s_cab1[j];
        #pragma unroll
        for (int c = 0; c < 64; ++c) acc = fmaf(a[c], s_caw1[j * 64 + c], acc);
        t1[j] = fmaxf(acc, 0.f);
    }
    #pragma unroll
    for (int c = 0; c < 64; ++c) {
        float att = s_cab2[c];
        #pragma unroll
        for (int j = 0; j < 16; ++j) att = fmaf(t1[j], s_caw2[c * 16 + j], att);
        a[c] *= att;   // x = x_norm * att
    }

    // spatial path: 64 -> 16 (bn1,relu) -> 64 (bn2) -> sigmoid gate
    float h1[16];
    #pragma unroll
    for (int j = 0; j < 16; ++j) {
        float acc = s_cb1[j];
        #pragma unroll
        for (int c = 0; c < 64; ++c) acc = fmaf(a[c], s_cw1[j * 64 + c], acc);
        h1[j] = fmaxf(acc * s_bn1s[j] + s_bn1o[j], 0.f);
    }
    #pragma unroll
    for (int c = 0; c < 64; ++c) {
        float h2 = s_cb2[c];
        #pragma unroll
        for (int j = 0; j < 16; ++j) h2 = fmaf(h1[j], s_cw2[c * 16 + j], h2);
        h2 = h2 * s_bn2s[c] + s_bn2o[c];
        const float sig = 1.0f / (1.0f + __expf(-h2));
        const float outv = a[c] * sig;
        // w is re-read by the GEMM: keep regular (RT) store
        w_out[(size_t)row * 64 + c] =
            (outv - s_revb[c]) / (s_revw[c] + EPS * EPS) * sd + mu;
    }
}

// ---------------------------------------------------------------------------
// Kernel 4: recon = data @ w^T  (M=N=8192, K=64, fp32 WMMA 16x16x4)
// Block tile 128(M) x 64(N), 8 waves; each wave: 32x32 via 2x2 accumulators.
// LDS stride 68 -> fragment ds_load_b64 hits all 64 banks exactly once.
// recon stores are non-temporal: 268MB stream written once, never re-read;
// keeps the 2MB data/w operand sets hot in the 192MB L2 across 8192 blocks.
// ---------------------------------------------------------------------------
__global__ __launch_bounds__(256) void k_gemm(
    const float* __restrict__ data, const float* __restrict__ w,
    float* __restrict__ recon)
{
    __shared__ float sA[128 * 68];
    __shared__ float sB[64 * 68];

    const int tid = threadIdx.x;
    const int blockM = blockIdx.y * 128;
    const int blockN = blockIdx.x * 64;

    // cooperative fill: A tile 128x64 (2048 float4), B tile 64x64 (1024 float4)
    {
        const float4* gA = (const float4*)(data + (size_t)blockM * 64);
        #pragma unroll
        for (int i = 0; i < 8; ++i) {
            const int id = tid + i * 256;
            const int r = id >> 4, c4 = id & 15;
            *(float4*)&sA[r * 68 + c4 * 4] = gA[id];
        }
        const float4* gB = (const float4*)(w + (size_t)blockN * 64);
        #pragma unroll
        for (int i = 0; i < 4; ++i) {
            const int id = tid + i * 256;
            const int r = id >> 4, c4 = id & 15;
            *(float4*)&sB[r * 68 + c4 * 4] = gB[id];
        }
    }
    __syncthreads();

    const int wave = tid >> 5;
    const int lane = tid & 31;
    const int wm = wave >> 1;          // 0..3 -> 32 rows each
    const int wn = wave & 1;           // 0..1 -> 32 cols each
    const int half = lane >> 4;        // K pair select
    const int lr = lane & 15;          // row/col within 16

    v8f acc00 = {}, acc01 = {}, acc10 = {}, acc11 = {};
    const int aRow0 = wm * 32 + lr;
    const int bRow0 = wn * 32 + lr;

    #pragma unroll
    for (int k0 = 0; k0 < 64; k0 += 4) {
        const int kk = k0 + 2 * half;
        const v2f a0 = *(const v2f*)&sA[(aRow0     ) * 68 + kk];
        const v2f a1 = *(const v2f*)&sA[(aRow0 + 16) * 68 + kk];
        const v2f b0 = *(const v2f*)&sB[(bRow0     ) * 68 + kk];
        const v2f b1 = *(const v2f*)&sB[(bRow0 + 16) * 68 + kk];
        acc00 = __builtin_amdgcn_wmma_f32_16x16x4_f32(false, a0, false, b0, (short)0, acc00, false, false);
        acc01 = __builtin_amdgcn_wmma_f32_16x16x4_f32(false, a0, false, b1, (short)0, acc01, false, false);
        acc10 = __builtin_amdgcn_wmma_f32_16x16x4_f32(false, a1, false, b0, (short)0, acc10, false, false);
        acc11 = __builtin_amdgcn_wmma_f32_16x16x4_f32(false, a1, false, b1, (short)0, acc11, false, false);
    }

    // D layout: VGPR v, lanes 0-15 -> M=v, lanes 16-31 -> M=v+8; N = lane%16
    const int colBase = blockN + wn * 32 + lr;
    #pragma unroll
    for (int v = 0; v < 8; ++v) {
        const int m = v + 8 * half;
        const size_t r0 = (size_t)(blockM + wm * 32 + m);
        const size_t r1 = (size_t)(blockM + wm * 32 + 16 + m);
        __builtin_nontemporal_store(acc00[v], &recon[r0 * B_DIM + colBase +  0]);
        __builtin_nontemporal_store(acc01[v], &recon[r0 * B_DIM + colBase + 16]);
        __builtin_nontemporal_store(acc10[v], &recon[r1 * B_DIM + colBase +  0]);
        __builtin_nontemporal_store(acc11[v], &recon[r1 * B_DIM + colBase + 16]);
    }
}

// ---------------------------------------------------------------------------
extern "C" void kernel_launch(void* const* d_in, const int* in_sizes, int n_in,
                              void* d_out, int out_size, void* d_ws, size_t ws_size,
                              hipStream_t stream)
{
    (void)in_sizes; (void)n_in; (void)out_size; (void)ws_size;

    const float* pred    = (const float*)d_in[0];
    const float* imfs    = (const float*)d_in[1];
    const float* rev_w   = (const float*)d_in[2];
    const float* rev_b   = (const float*)d_in[3];
    const float* ca_w1   = (const float*)d_in[4];
    const float* ca_b1   = (const float*)d_in[5];
    const float* ca_w2   = (const float*)d_in[6];
    const float* ca_b2   = (const float*)d_in[7];
    const float* conv1_w = (const float*)d_in[8];
    const float* conv1_b = (const float*)d_in[9];
    const float* bn1_g   = (const float*)d_in[10];
    const float* bn1_b   = (const float*)d_in[11];
    const float* bn1_m   = (const float*)d_in[12];
    const float* bn1_v   = (const float*)d_in[13];
    const float* conv2_w = (const float*)d_in[14];
    const float* conv2_b = (const float*)d_in[15];
    const float* bn2_g   = (const float*)d_in[16];
    const float* bn2_b   = (const float*)d_in[17];
    const float* bn2_m   = (const float*)d_in[18];
    const float* bn2_v   = (const float*)d_in[19];

    float* out       = (float*)d_out;
    float* recon     = out;                                    // (B,B)
    float* w_out     = recon + (size_t)B_DIM * B_DIM;          // (B,C)
    float* data_out  = w_out + (size_t)B_DIM * C_DIM;          // (B,C)
    float* xnorm_out = data_out + (size_t)B_DIM * C_DIM;       // (B,C)

    float* partials = (float*)d_ws;        // 2048 * 2 floats, fully rewritten
    float* stats    = partials + 4096;     // mu, std

    k_build_data<<<2048, 256, 0, stream>>>(pred, imfs, data_out, partials);
    k_stats<<<1, 256, 0, stream>>>(partials, stats);
    k_rows<<<B_DIM / 256, 256, 0, stream>>>(
        data_out, stats, rev_w, rev_b, ca_w1, ca_b1, ca_w2, ca_b2,
        conv1_w, conv1_b, bn1_g, bn1_b, bn1_m, bn1_v,
        conv2_w, conv2_b, bn2_g, bn2_b, bn2_m, bn2_v,
        w_out, xnorm_out);
    dim3 grid(B_DIM / 64, B_DIM / 128);
    k_gemm<<<grid, 256, 0, stream>>>(data_out, w_out, recon);
}